// GIAT_14276471292400
// MI455X (gfx1250) — compile-verified
//
#include <hip/hip_runtime.h>
#include <math.h>

#define N_NODES   50000
#define N_EDGES   800000
#define N_FEAT    128
#define DIM       100
#define HEADS     4
#define NUM_GRAPHS 128

typedef __attribute__((ext_vector_type(2))) float v2f;
typedef __attribute__((ext_vector_type(8))) float v8f;

// ---------------- workspace layout (in floats) ----------------
#define OFF_H0    0L                          // [N,128] x+agg
#define OFF_HID1  (OFF_H0   + 6400000L)       // [N,208]
#define OFF_H     (OFF_HID1 + 10400000L)      // [N,112]
#define OFF_XP    (OFF_H    + 5600000L)       // [N,400]
#define OFF_OUT   (OFF_XP   + 20000000L)      // [N,400]
#define OFF_AS    (OFF_OUT  + 20000000L)      // [N,4]
#define OFF_AD    (OFF_AS   + 200000L)        // [N,4]
#define OFF_M     (OFF_AD   + 200000L)        // [N,4]
#define OFF_DEN   (OFF_M    + 200000L)        // [N,4]
#define OFF_EBUF  (OFF_DEN  + 200000L)        // [(E+N),4]
#define OFF_STATS (OFF_EBUF + 3400000L)       // 256 (sum @0, sumsq @128)
#define OFF_SCSH  (OFF_STATS+ 256L)           // 256 (scale @0, shift @128)
#define OFF_G     (OFF_SCSH + 256L)           // [128,400]
#define OFF_G1    (OFF_G    + 51200L)         // [128,304]
#define OFF_G2    (OFF_G1   + 38912L)         // [128,208]
#define OFF_G3    (OFF_G2   + 26624L)         // [128,112]
#define OFF_WP1   (OFF_G3   + 14336L)         // [128,208]
#define OFF_B1P   (OFF_WP1  + 26624L)         // 208
#define OFF_WP2   (OFF_B1P  + 208L)           // [208,112]
#define OFF_B2P   (OFF_WP2  + 23296L)         // 112
#define OFF_WF1   (OFF_B2P  + 112L)           // [400,304]
#define OFF_BF1   (OFF_WF1  + 121600L)        // 304
#define OFF_WF2   (OFF_BF1  + 304L)           // [304,208]
#define OFF_BF2   (OFF_WF2  + 63232L)         // 208
#define OFF_WF3   (OFF_BF2  + 208L)           // [208,112]
#define OFF_BF3   (OFF_WF3  + 23296L)         // 112

// ---------------- small utility kernels ----------------
__global__ void k_fill(float* p, float v, long n) {
  long i = blockIdx.x * (long)blockDim.x + threadIdx.x;
  if (i < n) p[i] = v;
}

// zero-padded copy: dst[r*dcols+c] = (r<srows && c<scols) ? src[r*scols+c] : 0
__global__ void k_padcopy(float* dst, const float* src, int drows, int dcols,
                          int srows, int scols) {
  long i = blockIdx.x * (long)blockDim.x + threadIdx.x;
  long tot = (long)drows * dcols;
  if (i >= tot) return;
  int r = (int)(i / dcols), c = (int)(i % dcols);
  dst[i] = (r < srows && c < scols) ? src[(long)r * scols + c] : 0.0f;
}

// ---------------- GIN aggregation: h0 += scatter_add(x[src] -> dst) --------
// one wave per edge; each lane moves a float4 (32*4 = 128 feats)
__global__ void k_gin_agg(const float* __restrict__ x, const int* __restrict__ ei,
                          float* __restrict__ h0) {
  long wave = (blockIdx.x * (long)blockDim.x + threadIdx.x) >> 5;
  int lane = threadIdx.x & 31;
  if (wave >= N_EDGES) return;
  int s = ei[wave];
  int d = ei[N_EDGES + wave];
  const float4 v = ((const float4*)(x + (long)s * N_FEAT))[lane];
  float* dst = h0 + (long)d * N_FEAT + lane * 4;
  atomicAdd(dst + 0, v.x);
  atomicAdd(dst + 1, v.y);
  atomicAdd(dst + 2, v.z);
  atomicAdd(dst + 3, v.w);
}

// ---------------- fp32 WMMA strip GEMM ----------------
// One wave owns a 16-row strip and NT consecutive 16-col tiles: the A
// fragment is loaded ONCE per k-step and reused across all NT WMMAs, so A
// streams from HBM exactly once per GEMM. B fragments are batched in groups
// of 4 (8 live VGPRs) -> load clause + 4 WMMAs, without blowing the VGPR
// budget (accumulators stay resident; no scratch spills). K loop kept
// non-pipelined (#pragma unroll 1) to bound register pressure.
// All buffers padded to 16-col multiples -> full tiles, EXEC all-ones.
template <int NT>
__global__ __launch_bounds__(256)
void k_gemm_strip(const float* __restrict__ A, int lda,
                  const float* __restrict__ B, int ldb,
                  float* __restrict__ C, int ldc,
                  const float* __restrict__ bias,
                  int K, int mtiles, int do_relu, int col0) {
  int wid = (int)((blockIdx.x * (long)blockDim.x + threadIdx.x) >> 5);
  if (wid >= mtiles) return;               // wave-uniform guard
  int lane = threadIdx.x & 31;
  int half = lane >> 4;                    // 0: K pair {k,k+1}, 1: {k+2,k+3}
  int l16  = lane & 15;

  const float* arow = A + (long)(wid * 16 + l16) * lda;
  const float* Bp   = B + col0 + l16;

  v8f acc[NT];
#pragma unroll
  for (int t = 0; t < NT; t++) acc[t] = (v8f){};

#pragma unroll 1
  for (int k = 0; k < K; k += 4) {
    int ka = k + half * 2;
    float2 av = *(const float2*)(arow + ka);   // contiguous in K
    v2f a; a.x = av.x; a.y = av.y;
    const float* b0 = Bp + (long)ka * ldb;
    const float* b1 = Bp + (long)(ka + 1) * ldb;
#pragma unroll
    for (int t0 = 0; t0 < NT; t0 += 4) {       // groups of 4 col tiles
      v2f bf[4];
#pragma unroll
      for (int j = 0; j < 4; j++) {
        int t = t0 + j;
        if (t < NT) { bf[j].x = b0[t * 16]; bf[j].y = b1[t * 16]; }
      }
#pragma unroll
      for (int j = 0; j < 4; j++) {
        int t = t0 + j;
        if (t < NT)
          acc[t] = __builtin_amdgcn_wmma_f32_16x16x4_f32(false, a, false, bf[j],
                                                         (short)0, acc[t], false, false);
      }
    }
  }

#pragma unroll
  for (int t = 0; t < NT; t++) {
    int col = col0 + t * 16 + l16;
    float bv = bias ? bias[col] : 0.0f;
#pragma unroll
    for (int v = 0; v < 8; v++) {
      int row = wid * 16 + v + half * 8;       // C/D layout: VGPR v -> M=v (+8)
      float val = acc[t][v] + bv;
      if (do_relu) val = fmaxf(val, 0.0f);
      C[(long)row * ldc + col] = val;
    }
  }
}

// ---------------- BatchNorm stats / apply ----------------
__global__ void k_bn_stats(const float* __restrict__ h, float* __restrict__ stats) {
  __shared__ float ssum[256], ssq[256];
  int ch = blockIdx.x;                       // 0..99
  float s = 0.0f, q = 0.0f;
  for (int r = threadIdx.x; r < N_NODES; r += 256) {
    float v = h[(long)r * 112 + ch];
    s += v; q += v * v;
  }
  ssum[threadIdx.x] = s; ssq[threadIdx.x] = q;
  __syncthreads();
  for (int off = 128; off > 0; off >>= 1) {
    if ((int)threadIdx.x < off) {
      ssum[threadIdx.x] += ssum[threadIdx.x + off];
      ssq[threadIdx.x]  += ssq[threadIdx.x + off];
    }
    __syncthreads();
  }
  if (threadIdx.x == 0) { stats[ch] = ssum[0]; stats[128 + ch] = ssq[0]; }
}

__global__ void k_bn_scale(const float* stats, const float* gamma,
                           const float* beta, float* scsh) {
  int ch = threadIdx.x;
  if (ch < DIM) {
    float mean = stats[ch] * (1.0f / N_NODES);
    float var  = stats[128 + ch] * (1.0f / N_NODES) - mean * mean;
    float sc = gamma[ch] * rsqrtf(var + 1e-5f);
    scsh[ch] = sc;
    scsh[128 + ch] = beta[ch] - mean * sc;
  }
}

__global__ void k_bn_apply(float* h, const float* __restrict__ scsh) {
  long i = blockIdx.x * (long)blockDim.x + threadIdx.x;
  if (i >= (long)N_NODES * DIM) return;
  int r = (int)(i / DIM), ch = (int)(i % DIM);
  float* p = h + (long)r * 112 + ch;
  *p = (*p) * scsh[ch] + scsh[128 + ch];
}

// ---------------- GAT attention ----------------
__global__ void k_attn(const float* __restrict__ xp, const float* __restrict__ att_src,
                       const float* __restrict__ att_dst, float* a_s, float* a_d) {
  long i = blockIdx.x * (long)blockDim.x + threadIdx.x;   // n*4 + head
  if (i >= (long)N_NODES * HEADS) return;
  int n = (int)(i >> 2), hh = (int)(i & 3);
  const float* v  = xp + (long)n * 400 + hh * DIM;
  const float* as = att_src + hh * DIM;
  const float* ad = att_dst + hh * DIM;
  float ss = 0.0f, dd = 0.0f;
  for (int k = 0; k < DIM; k++) { float xv = v[k]; ss += xv * as[k]; dd += xv * ad[k]; }
  a_s[i] = ss; a_d[i] = dd;
}

__device__ inline void atomicMaxFloat(float* addr, float val) {
  if (val >= 0.0f) atomicMax((int*)addr, __float_as_int(val));
  else             atomicMin((unsigned int*)addr, (unsigned int)__float_as_int(val));
}

__global__ void k_edge_max(const int* __restrict__ ei, const float* __restrict__ a_s,
                           const float* __restrict__ a_d, float* ebuf, float* m) {
  long i = blockIdx.x * (long)blockDim.x + threadIdx.x;   // edge*4 + head
  if (i >= (long)(N_EDGES + N_NODES) * HEADS) return;
  long e = i >> 2; int hh = (int)(i & 3);
  int s, d;
  if (e < N_EDGES) { s = ei[e]; d = ei[N_EDGES + e]; }
  else             { s = d = (int)(e - N_EDGES); }        // self-loops
  float v = a_s[(long)s * 4 + hh] + a_d[(long)d * 4 + hh];
  v = (v > 0.0f) ? v : 0.2f * v;                          // leaky_relu
  ebuf[i] = v;
  atomicMaxFloat(&m[(long)d * 4 + hh], v);
}

__global__ void k_edge_exp(const int* __restrict__ ei, float* ebuf,
                           const float* __restrict__ m, float* denom) {
  long i = blockIdx.x * (long)blockDim.x + threadIdx.x;
  if (i >= (long)(N_EDGES + N_NODES) * HEADS) return;
  long e = i >> 2; int hh = (int)(i & 3);
  int d = (e < N_EDGES) ? ei[N_EDGES + e] : (int)(e - N_EDGES);
  float w = expf(ebuf[i] - m[(long)d * 4 + hh]);
  ebuf[i] = w;
  atomicAdd(&denom[(long)d * 4 + hh], w);
}

// one wave per edge: out[d][0..399] += xp[s][0..399] * alpha[head(k)]
__global__ void k_edge_scatter(const int* __restrict__ ei, const float* __restrict__ ebuf,
                               const float* __restrict__ denom,
                               const float* __restrict__ xp, float* __restrict__ out) {
  long wave = (blockIdx.x * (long)blockDim.x + threadIdx.x) >> 5;
  int lane = threadIdx.x & 31;
  if (wave >= (long)(N_EDGES + N_NODES)) return;
  int s, d;
  if (wave < N_EDGES) { s = ei[wave]; d = ei[N_EDGES + wave]; }
  else                { s = d = (int)(wave - N_EDGES); }
  float alpha[HEADS];
#pragma unroll
  for (int hh = 0; hh < HEADS; hh++)
    alpha[hh] = ebuf[wave * 4 + hh] / denom[(long)d * 4 + hh];
  const float* src = xp + (long)s * 400;
  float* dst = out + (long)d * 400;
  for (int k = lane; k < HEADS * DIM; k += 32) {
    int hh = k / DIM;
    atomicAdd(&dst[k], src[k] * alpha[hh]);
  }
}

// ---------------- global add pool (+ GAT bias) ----------------
__global__ void k_pool(const float* __restrict__ out, const float* __restrict__ bg,
                       const int* __restrict__ batch, float* __restrict__ g) {
  long i = blockIdx.x * (long)blockDim.x + threadIdx.x;
  if (i >= (long)N_NODES * 400) return;
  int n = (int)(i / 400), f = (int)(i % 400);
  atomicAdd(&g[(long)batch[n] * 400 + f], out[i] + bg[f]);
}

// ---------------- final 100 -> 1 projection ----------------
__global__ void k_final(const float* __restrict__ g3, const float* __restrict__ Wf4,
                        const float* __restrict__ bf4, float* __restrict__ o) {
  int gi = blockIdx.x * blockDim.x + threadIdx.x;
  if (gi >= NUM_GRAPHS) return;
  float acc = bf4[0];
  for (int k = 0; k < DIM; k++) acc += g3[(long)gi * 112 + k] * Wf4[k];
  o[gi] = acc;
}

static inline int cdiv(long a, long b) { return (int)((a + b - 1) / b); }

extern "C" void kernel_launch(void* const* d_in, const int* in_sizes, int n_in,
                              void* d_out, int out_size, void* d_ws, size_t ws_size,
                              hipStream_t stream) {
  const float* x       = (const float*)d_in[0];
  const int*   ei      = (const int*)  d_in[1];
  const int*   batch   = (const int*)  d_in[2];
  const float* W1      = (const float*)d_in[3];
  const float* b1      = (const float*)d_in[4];
  const float* W2      = (const float*)d_in[5];
  const float* b2      = (const float*)d_in[6];
  const float* gamma   = (const float*)d_in[7];
  const float* beta    = (const float*)d_in[8];
  const float* Wg      = (const float*)d_in[9];
  const float* att_src = (const float*)d_in[10];
  const float* att_dst = (const float*)d_in[11];
  const float* bg      = (const float*)d_in[12];
  const float* Wf1     = (const float*)d_in[13];
  const float* bf1     = (const float*)d_in[14];
  const float* Wf2     = (const float*)d_in[15];
  const float* bf2     = (const float*)d_in[16];
  const float* Wf3     = (const float*)d_in[17];
  const float* bf3     = (const float*)d_in[18];
  const float* Wf4     = (const float*)d_in[19];
  const float* bf4     = (const float*)d_in[20];
  float* o = (float*)d_out;
  float* ws = (float*)d_ws;

  float* h0   = ws + OFF_H0;
  float* hid1 = ws + OFF_HID1;
  float* h    = ws + OFF_H;
  float* xp   = ws + OFF_XP;
  float* outb = ws + OFF_OUT;
  float* a_s  = ws + OFF_AS;
  float* a_d  = ws + OFF_AD;
  float* m    = ws + OFF_M;
  float* den  = ws + OFF_DEN;
  float* ebuf = ws + OFF_EBUF;
  float* stats= ws + OFF_STATS;
  float* scsh = ws + OFF_SCSH;
  float* g    = ws + OFF_G;
  float* g1   = ws + OFF_G1;
  float* g2   = ws + OFF_G2;
  float* g3   = ws + OFF_G3;
  float* Wp1  = ws + OFF_WP1;  float* b1p = ws + OFF_B1P;
  float* Wp2  = ws + OFF_WP2;  float* b2p = ws + OFF_B2P;
  float* Wpf1 = ws + OFF_WF1;  float* bf1p = ws + OFF_BF1;
  float* Wpf2 = ws + OFF_WF2;  float* bf2p = ws + OFF_BF2;
  float* Wpf3 = ws + OFF_WF3;  float* bf3p = ws + OFF_BF3;

  // --- init accumulators ---
  hipMemsetAsync(outb, 0, (size_t)N_NODES * 400 * 4, stream);
  hipMemsetAsync(den,  0, (size_t)N_NODES * 4 * 4, stream);
  hipMemsetAsync(g,    0, (size_t)NUM_GRAPHS * 400 * 4, stream);
  k_fill<<<cdiv((long)N_NODES * 4, 256), 256, 0, stream>>>(m, -INFINITY, (long)N_NODES * 4);

  // --- pad weights/biases into WMMA-friendly shapes ---
  k_padcopy<<<cdiv(128L*208, 256), 256, 0, stream>>>(Wp1, W1, 128, 208, 128, 200);
  k_padcopy<<<1, 256, 0, stream>>>(b1p, b1, 1, 208, 1, 200);
  k_padcopy<<<cdiv(208L*112, 256), 256, 0, stream>>>(Wp2, W2, 208, 112, 200, 100);
  k_padcopy<<<1, 256, 0, stream>>>(b2p, b2, 1, 112, 1, 100);
  k_padcopy<<<cdiv(400L*304, 256), 256, 0, stream>>>(Wpf1, Wf1, 400, 304, 400, 300);
  k_padcopy<<<2, 256, 0, stream>>>(bf1p, bf1, 1, 304, 1, 300);
  k_padcopy<<<cdiv(304L*208, 256), 256, 0, stream>>>(Wpf2, Wf2, 304, 208, 300, 200);
  k_padcopy<<<1, 256, 0, stream>>>(bf2p, bf2, 1, 208, 1, 200);
  k_padcopy<<<cdiv(208L*112, 256), 256, 0, stream>>>(Wpf3, Wf3, 208, 112, 200, 100);
  k_padcopy<<<1, 256, 0, stream>>>(bf3p, bf3, 1, 112, 1, 100);

  // --- GIN: h0 = x + scatter_add(x[src] -> dst) ---
  hipMemcpyAsync(h0, x, (size_t)N_NODES * N_FEAT * 4, hipMemcpyDeviceToDevice, stream);
  k_gin_agg<<<N_EDGES / 8, 256, 0, stream>>>(x, ei, h0);

  // --- GIN MLP (fp32 WMMA strips; A streamed once) ---
  int mstrips = N_NODES / 16;  // 3125
  k_gemm_strip<13><<<cdiv(mstrips, 8), 256, 0, stream>>>(h0, 128, Wp1, 208, hid1, 208,
                                                         b1p, 128, mstrips, 1, 0);
  k_gemm_strip<7><<<cdiv(mstrips, 8), 256, 0, stream>>>(hid1, 208, Wp2, 112, h, 112,
                                                        b2p, 208, mstrips, 1, 0);

  // --- BatchNorm (training batch stats) ---
  k_bn_stats<<<DIM, 256, 0, stream>>>(h, stats);
  k_bn_scale<<<1, 128, 0, stream>>>(stats, gamma, beta, scsh);
  k_bn_apply<<<cdiv((long)N_NODES * DIM, 256), 256, 0, stream>>>(h, scsh);

  // --- GAT linear: xp = h @ Wg  (K=100, 400 cols = 25 tiles -> 13+12 strips) ---
  k_gemm_strip<13><<<cdiv(mstrips, 8), 256, 0, stream>>>(h, 112, Wg, 400, xp, 400,
                                                         nullptr, 100, mstrips, 0, 0);
  k_gemm_strip<12><<<cdiv(mstrips, 8), 256, 0, stream>>>(h, 112, Wg, 400, xp, 400,
                                                         nullptr, 100, mstrips, 0, 208);

  // --- GAT attention: coefficients, segment softmax, weighted scatter ---
  k_attn<<<cdiv((long)N_NODES * HEADS, 256), 256, 0, stream>>>(xp, att_src, att_dst, a_s, a_d);
  k_edge_max<<<cdiv((long)(N_EDGES + N_NODES) * HEADS, 256), 256, 0, stream>>>(ei, a_s, a_d, ebuf, m);
  k_edge_exp<<<cdiv((long)(N_EDGES + N_NODES) * HEADS, 256), 256, 0, stream>>>(ei, ebuf, m, den);
  k_edge_scatter<<<cdiv((long)(N_EDGES + N_NODES) * 32, 256), 256, 0, stream>>>(ei, ebuf, den, xp, outb);

  // --- global_add_pool (+bg) ---
  k_pool<<<cdiv((long)N_NODES * 400, 256), 256, 0, stream>>>(outb, bg, batch, g);

  // --- MLP head (fp32 WMMA strips, M=128 -> 8 row strips) ---
  k_gemm_strip<19><<<1, 256, 0, stream>>>(g, 400, Wpf1, 304, g1, 304, bf1p, 400, 8, 1, 0);
  k_gemm_strip<13><<<1, 256, 0, stream>>>(g1, 304, Wpf2, 208, g2, 208, bf2p, 304, 8, 1, 0);
  k_gemm_strip<7><<<1, 256, 0, stream>>>(g2, 208, Wpf3, 112, g3, 112, bf3p, 208, 8, 1, 0);
  k_final<<<1, 128, 0, stream>>>(g3, Wf4, bf4, o);
}